// GraphAttentionLayer_48112223649840
// MI455X (gfx1250) — compile-verified
//
#include <hip/hip_runtime.h>

typedef float v2f __attribute__((ext_vector_type(2)));
typedef float v8f __attribute__((ext_vector_type(8)));
typedef unsigned int u32x4 __attribute__((ext_vector_type(4)));
typedef unsigned int u32x8 __attribute__((ext_vector_type(8)));

#define N_NODES 8192
#define F_IN    128
#define F_OUT   64
#define LRELU_ALPHA 0.2f

// Tensor Data Mover: one-instruction 2-D tile DMA, tracked by TENSORcnt.
// 2-D tensors: descriptor groups 0 (4 SGPRs) + 1 (8 SGPRs), groups 2/3 NULL.
#define TDM_LOAD_TO_LDS(g0, g1)                                         \
  asm volatile("tensor_load_to_lds %0, %1" :: "s"(g0), "s"(g1) : "memory")
#define WAIT_TENSORCNT_1() asm volatile("s_wait_tensorcnt 0x1" ::: "memory")
#define WAIT_TENSORCNT_0() asm volatile("s_wait_tensorcnt 0x0" ::: "memory")

// ---------------------------------------------------------------------------
// Kernel 1: wh = x @ w  via V_WMMA_F32_16X16X4_F32. One wave per 16-row tile.
// Also emits whPack[k/2][n][2] (pairs of consecutive wh rows interleaved) so
// kernel 3's B-fragments become single aligned b64 loads.
// A layout (16x4 f32): lane l holds M = l%16, VGPR v holds K = v + 2*(l/16).
// B layout (4x16 f32): lane l holds N = l%16, VGPR v holds K = v + 2*(l/16).
// C layout (16x16 f32): lane l, VGPR v holds C[M = v + 8*(l/16)][N = l%16].
// ---------------------------------------------------------------------------
__global__ __launch_bounds__(32)
void gat_xw_wmma(const float* __restrict__ x, const float* __restrict__ w,
                 float* __restrict__ wh, float* __restrict__ whPack) {
  const int lane = threadIdx.x & 31;
  const int g    = lane >> 4;
  const int ln   = lane & 15;
  const int rowBase = blockIdx.x * 16;

  v8f acc[4] = {};
  const float* xrow = x + (long long)(rowBase + ln) * F_IN;

  for (int k = 0; k < F_IN; k += 4) {
    v2f afrag = *(const v2f*)(xrow + k + 2 * g);
#pragma unroll
    for (int t = 0; t < 4; ++t) {
      v2f bfrag;
      bfrag.x = w[(k + 0 + 2 * g) * F_OUT + t * 16 + ln];
      bfrag.y = w[(k + 1 + 2 * g) * F_OUT + t * 16 + ln];
      acc[t] = __builtin_amdgcn_wmma_f32_16x16x4_f32(
          false, afrag, false, bfrag, (short)0, acc[t], false, false);
    }
  }
#pragma unroll
  for (int t = 0; t < 4; ++t)
#pragma unroll
    for (int v = 0; v < 8; ++v) {
      const int m = v + 8 * g;
      const int n = t * 16 + ln;
      const float val = acc[t][v];
      wh[(long long)(rowBase + m) * F_OUT + n] = val;
      whPack[(long long)((rowBase + m) >> 1) * (2 * F_OUT) + n * 2 + (m & 1)] = val;
    }
}

// ---------------------------------------------------------------------------
// Kernel 2: src[i] = wh[i,:] @ a[:64], dst[i] = wh[i,:] @ a[64:]
// ---------------------------------------------------------------------------
__global__ __launch_bounds__(256)
void gat_srcdst(const float* __restrict__ wh, const float* __restrict__ a,
                float* __restrict__ src, float* __restrict__ dst) {
  __shared__ float reds[256];
  __shared__ float redd[256];
  const int tid = threadIdx.x;
  const int row = blockIdx.x * 4 + (tid >> 6);
  const int c   = tid & 63;
  const float whv = wh[(long long)row * F_OUT + c];
  reds[tid] = whv * a[c];
  redd[tid] = whv * a[F_OUT + c];
  __syncthreads();
#pragma unroll
  for (int s = 32; s > 0; s >>= 1) {
    if (c < s) { reds[tid] += reds[tid + s]; redd[tid] += redd[tid + s]; }
    __syncthreads();
  }
  if (c == 0) { src[row] = reds[tid]; dst[row] = redd[tid]; }
}

// ---------------------------------------------------------------------------
// Kernel 3: fused masked-softmax attention + attn@wh + elu.
// One block (8 waves) per 16-row tile; each wave owns 1024 j-columns in
// 32-column chunks. Each 16x32-int adj tile is fetched by the Tensor Data
// Mover (tensor_load_to_lds, TENSORcnt) into a per-wave LDS double buffer,
// then consumed as ds_load_b64 in the WMMA A-fragment pattern.
// exp_term(i,j) = adj>0 ? exp(lrelu(src[i]+dst[j])) : exp(0)=1
// out = elu( (sum_j exp_term * wh[j,:]) / sum_j exp_term )
// ---------------------------------------------------------------------------
__global__ __launch_bounds__(256)
void gat_attn_wmma(const int* __restrict__ adj, const float* __restrict__ whPack,
                   const float* __restrict__ src, const float* __restrict__ dst,
                   float* __restrict__ out) {
  __shared__ int   adjTiles[8 * 2 * 16 * 32] __attribute__((aligned(16))); // 32 KB
  __shared__ float ldsPart[8 * 16 * F_OUT];                                // 32 KB
  __shared__ float ldsLSum[8 * 32];
  __shared__ float ldsRS[16];

  const int tid  = threadIdx.x;
  const int lane = tid & 31;
  const int g    = lane >> 4;
  const int ln   = lane & 15;
  const int rowBase = blockIdx.x * 16;
  // Wave id, forced uniform so TDM descriptor fields live in SGPRs cleanly.
  const int waveU = __builtin_amdgcn_readfirstlane(tid >> 5);

  const float srcv = src[rowBase + ln];

  const int JPW = N_NODES / 8;          // 1024 columns per wave
  const int j0  = waveU * JPW;
  const int NCH = JPW / 32;             // 32 chunks of 32 columns

  // ---- TDM descriptor (D#) -------------------------------------------------
  // Group1 (constant): data_size=4B; tensor 8192x8192; tile 32x16; stride 8192.
  u32x8 g1;
  g1[0] = 2u << 16;                                   // data_size code 2 = 4B
  g1[1] = ((unsigned)N_NODES & 0xFFFFu) << 16;        // tensor_dim0[15:0]
  g1[2] = ((unsigned)N_NODES >> 16) |                 // tensor_dim0[31:16]
          (((unsigned)N_NODES & 0xFFFFu) << 16);      // tensor_dim1[15:0]
  g1[3] = ((unsigned)N_NODES >> 16) | (32u << 16);    // dim1[31:16] | tile_dim0
  g1[4] = 16u;                                        // tile_dim1 (tile_dim2=0)
  g1[5] = (unsigned)N_NODES;                          // tensor_dim0_stride lo32
  g1[6] = 0u;                                         // stride hi | dim1_stride lo
  g1[7] = 0u;

  const unsigned ldsAdjBase =
      (unsigned)(size_t)(void*)adjTiles + (unsigned)waveU * 4096u;
  unsigned long long ga =
      (unsigned long long)(size_t)(adj + (long long)rowBase * N_NODES + j0);

  v8f acc[4] = {};
  float lsum = 0.f;

  // prologue: DMA chunk 0 into buffer 0
  {
    u32x4 g0;
    g0[0] = 1u;                                       // count=1 (valid user D#)
    g0[1] = ldsAdjBase;
    g0[2] = (unsigned)ga;
    g0[3] = (unsigned)(ga >> 32) | 0x80000000u;       // type=2 ("image")
    TDM_LOAD_TO_LDS(g0, g1);
  }

  int buf = 0;
  for (int c = 0; c < NCH; ++c) {
    const int jb = j0 + c * 32;
    if (c + 1 < NCH) {
      u32x4 g0;
      g0[0] = 1u;
      g0[1] = ldsAdjBase + (unsigned)((buf ^ 1) * 2048);
      g0[2] = (unsigned)(ga + 128ull);
      g0[3] = (unsigned)((ga + 128ull) >> 32) | 0x80000000u;
      TDM_LOAD_TO_LDS(g0, g1);
      WAIT_TENSORCNT_1();     // tensor ops complete in order: current buf ready
    } else {
      WAIT_TENSORCNT_0();
    }

    const int* abuf = adjTiles + waveU * 1024 + buf * 512;
#pragma unroll
    for (int s = 0; s < 8; ++s) {
      const int kc = 4 * s + 2 * g;           // even K offset within chunk
      const int jc = jb + kc;
      int2 ad = *(const int2*)(abuf + ln * 32 + kc);     // ds_load_b64
      v2f  dv = *(const v2f*)(dst + jc);
      float e0 = srcv + dv.x; e0 = e0 > 0.f ? e0 : LRELU_ALPHA * e0;
      float e1 = srcv + dv.y; e1 = e1 > 0.f ? e1 : LRELU_ALPHA * e1;
      e0 = ad.x > 0 ? __expf(e0) : 1.0f;
      e1 = ad.y > 0 ? __expf(e1) : 1.0f;
      lsum += e0 + e1;
      v2f afrag; afrag.x = e0; afrag.y = e1;  // E[M=ln][K=jc..jc+1]
      const float* wb = whPack + (long long)(jc >> 1) * (2 * F_OUT);
#pragma unroll
      for (int t = 0; t < 4; ++t) {
        v2f bfrag = *(const v2f*)(wb + (t * 16 + ln) * 2); // {wh[jc][n], wh[jc+1][n]}
        acc[t] = __builtin_amdgcn_wmma_f32_16x16x4_f32(
            false, afrag, false, bfrag, (short)0, acc[t], false, false);
      }
    }
    ga  += 128ull;            // advance 32 int32 columns
    buf ^= 1;
  }

  // Dump per-wave partials to LDS (C layout: M = v + 8g, N = 16t + ln).
  float* myPart = ldsPart + waveU * (16 * F_OUT);
#pragma unroll
  for (int t = 0; t < 4; ++t)
#pragma unroll
    for (int v = 0; v < 8; ++v)
      myPart[(v + 8 * g) * F_OUT + t * 16 + ln] = acc[t][v];
  ldsLSum[waveU * 32 + lane] = lsum;
  __syncthreads();

  // Row sums: lanes ln and ln+16 of every wave cover disjoint K of row ln.
  if (tid < 16) {
    float rs = 0.f;
#pragma unroll
    for (int w8 = 0; w8 < 8; ++w8)
      rs += ldsLSum[w8 * 32 + tid] + ldsLSum[w8 * 32 + tid + 16];
    ldsRS[tid] = rs;
  }
  __syncthreads();

  // Reduce 8 partial tiles, normalize, elu, store (coalesced in n).
#pragma unroll
  for (int p = 0; p < 4; ++p) {
    const int idx = tid + p * 256;      // idx = m*64 + n
    const int m = idx >> 6;
    float h = 0.f;
#pragma unroll
    for (int w8 = 0; w8 < 8; ++w8)
      h += ldsPart[w8 * (16 * F_OUT) + idx];
    h /= ldsRS[m];
    out[(long long)rowBase * F_OUT + idx] = h > 0.f ? h : expm1f(h);
  }
}

// ---------------------------------------------------------------------------
extern "C" void kernel_launch(void* const* d_in, const int* in_sizes, int n_in,
                              void* d_out, int out_size, void* d_ws, size_t ws_size,
                              hipStream_t stream) {
  const float* x   = (const float*)d_in[0];   // [8192,128]
  const int*   adj = (const int*)d_in[1];     // [8192,8192]
  const float* w   = (const float*)d_in[2];   // [128,64]
  const float* a   = (const float*)d_in[3];   // [128,1]
  float* out = (float*)d_out;                 // [8192,64]

  float* wh     = (float*)d_ws;                    // 8192*64 floats
  float* whPack = wh + (size_t)N_NODES * F_OUT;    // 8192*64 floats (paired)
  float* src    = whPack + (size_t)N_NODES * F_OUT;
  float* dst    = src + N_NODES;

  gat_xw_wmma  <<<N_NODES / 16, 32,  0, stream>>>(x, w, wh, whPack);
  gat_srcdst   <<<N_NODES / 4,  256, 0, stream>>>(wh, a, src, dst);
  gat_attn_wmma<<<N_NODES / 16, 256, 0, stream>>>(adj, whPack, src, dst, out);
}